// STFNConv_26465588478210
// MI455X (gfx1250) — compile-verified
//
#include <hip/hip_runtime.h>
#include <hip/hip_bf16.h>

// ---------------------------------------------------------------------------
// Spiking GAT layer for MI455X (gfx1250), wave32.
// Pipeline (all on `stream`):
//   K1  proj GEMM (fp32 WMMA 16x16x4) + attention logits a_src/a_dst
//   K2  zero segment accumulators (mkey/denom/cnt/agg)
//   K3  edge pass 1: segment-max via order-preserving u32 atomicMax + degree
//   K4  edge pass 2: exp-sum (denominator)
//   K5  edge pass 3: alpha-weighted scatter of h[src] (f32 atomics, 1 wave/edge)
//   K6  output GEMM (fp32 WMMA) + row mean/var norm + LIF threshold
// ---------------------------------------------------------------------------

typedef __attribute__((ext_vector_type(2))) float v2f;
typedef __attribute__((ext_vector_type(8))) float v8f;

#define NN 100000
#define NE 1600000
#define NEG_SLOPE 0.2f
#define EPS_N 1e-5f
#define TAU_INV 0.5f
#define V_TH 1.0f

// workspace layout (float offsets)
constexpr int OFF_H    = 0;         // [NN*64] projected features h
constexpr int OFF_ASRC = 6400000;   // [NN*4]
constexpr int OFF_ADST = 6800000;   // [NN*4]
constexpr int OFF_MKEY = 7200000;   // [NN*4]  (unsigned keys)
constexpr int OFF_DEN  = 7600000;   // [NN*4]
constexpr int OFF_CNT  = 8000000;   // [NN]
constexpr int OFF_AGG  = 8100000;   // [NN*64]
constexpr int OFF_END  = 14500000;

// order-preserving float <-> uint key (monotonic for atomicMax, init key = 0)
__device__ __forceinline__ unsigned fkey(float f) {
    unsigned b = __float_as_uint(f);
    return b ^ (unsigned)(((int)b >> 31) | 0x80000000);
}
__device__ __forceinline__ float funkey(unsigned k) {
    unsigned b = k ^ ((k & 0x80000000u) ? 0x80000000u : 0xFFFFFFFFu);
    return __uint_as_float(b);
}

// --------------------------- K1: proj + logits ------------------------------
// One wave computes a 16-row x 64-col strip of h = x @ W^T via 16 steps of
// v_wmma_f32_16x16x4_f32, A-frag reused across the 4 column tiles (= heads).
__global__ __launch_bounds__(256)
void k_proj_att(const float* __restrict__ x, const float* __restrict__ W,
                const float* __restrict__ att_src, const float* __restrict__ att_dst,
                float* __restrict__ hbuf, float* __restrict__ a_src,
                float* __restrict__ a_dst) {
    int wave = (blockIdx.x * blockDim.x + threadIdx.x) >> 5;
    int lane = threadIdx.x & 31;
    int r0 = wave * 16;
    if (r0 >= NN) return;                 // wave-uniform: EXEC stays all-ones
    int half = lane >> 4;                 // K sub-block select
    int l = lane & 15;                    // row (A) / col (B,D) index

    const float* xrow = x + (size_t)(r0 + l) * 64 + 2 * half;
    v8f acc[4] = {v8f{}, v8f{}, v8f{}, v8f{}};

    for (int k = 0; k < 64; k += 4) {
        float2 av = *(const float2*)(xrow + k);
        v2f a; a[0] = av.x; a[1] = av.y;
#pragma unroll
        for (int t = 0; t < 4; ++t) {
            // B[k'][o] = W[o][k'];  o = t*16 + l
            float2 bv = *(const float2*)(W + (size_t)(t * 16 + l) * 64 + k + 2 * half);
            v2f b; b[0] = bv.x; b[1] = bv.y;
            acc[t] = __builtin_amdgcn_wmma_f32_16x16x4_f32(
                false, a, false, b, (short)0, acc[t], false, false);
        }
    }

    // store h + per-head attention dots (head t == column tile t, width 16)
#pragma unroll
    for (int t = 0; t < 4; ++t) {
        float aws = att_src[t * 16 + l];
        float awd = att_dst[t * 16 + l];
#pragma unroll
        for (int r = 0; r < 8; ++r) {
            float val = acc[t][r];
            int row = r0 + half * 8 + r;
            hbuf[(size_t)row * 64 + t * 16 + l] = val;
            float ps = val * aws;
            float pd = val * awd;
#pragma unroll
            for (int m = 1; m < 16; m <<= 1) {   // reduce within 16-lane half
                ps += __shfl_xor(ps, m, 32);
                pd += __shfl_xor(pd, m, 32);
            }
            if (l == 0) {
                a_src[row * 4 + t] = ps;
                a_dst[row * 4 + t] = pd;
            }
        }
    }
}

// --------------------------- K2: init accumulators --------------------------
__global__ __launch_bounds__(256)
void k_init(float* __restrict__ ws) {
    int t = blockIdx.x * blockDim.x + threadIdx.x;
    int idx = OFF_MKEY + t * 4;
    if (idx < OFF_END) *(float4*)(ws + idx) = make_float4(0.f, 0.f, 0.f, 0.f);
}

// --------------------------- K3: segment max + degree -----------------------
__global__ __launch_bounds__(256)
void k_edge_max(const int* __restrict__ ei, const float* __restrict__ a_src,
                const float* __restrict__ a_dst, unsigned* __restrict__ mkey,
                float* __restrict__ cnt) {
    int t = blockIdx.x * blockDim.x + threadIdx.x;
    if (t >= NE * 4) return;
    int e = t >> 2, hd = t & 3;
    int src = ei[e], dst = ei[NE + e];
    float ev = a_src[src * 4 + hd] + a_dst[dst * 4 + hd];
    ev = ev >= 0.f ? ev : NEG_SLOPE * ev;
    atomicMax(mkey + dst * 4 + hd, fkey(ev));
    if (hd == 0) atomicAdd(cnt + dst, 1.0f);
}

// --------------------------- K4: softmax denominator ------------------------
__global__ __launch_bounds__(256)
void k_edge_den(const int* __restrict__ ei, const float* __restrict__ a_src,
                const float* __restrict__ a_dst, const unsigned* __restrict__ mkey,
                float* __restrict__ denom) {
    int t = blockIdx.x * blockDim.x + threadIdx.x;
    if (t >= NE * 4) return;
    int e = t >> 2, hd = t & 3;
    int src = ei[e], dst = ei[NE + e];
    float ev = a_src[src * 4 + hd] + a_dst[dst * 4 + hd];
    ev = ev >= 0.f ? ev : NEG_SLOPE * ev;
    float m = funkey(mkey[dst * 4 + hd]);
    atomicAdd(denom + dst * 4 + hd, __expf(ev - m));
}

// --------------------------- K5: weighted scatter ---------------------------
// One wave per edge: 32 lanes x float2 = 64 channels.
__global__ __launch_bounds__(256)
void k_edge_agg(const int* __restrict__ ei, const float* __restrict__ a_src,
                const float* __restrict__ a_dst, const unsigned* __restrict__ mkey,
                const float* __restrict__ denom, const float* __restrict__ hbuf,
                float* __restrict__ agg) {
    int t = blockIdx.x * blockDim.x + threadIdx.x;
    int e = t >> 5;
    if (e >= NE) return;
    int p = threadIdx.x & 31;               // channel pair index
    int sv = 0, dv = 0;
    if (p == 0) { sv = ei[e]; dv = ei[NE + e]; }
    int src = __shfl(sv, 0, 32);
    int dst = __shfl(dv, 0, 32);
    int hd = p >> 3;                        // head for channels 2p,2p+1
    float ev = a_src[src * 4 + hd] + a_dst[dst * 4 + hd];
    ev = ev >= 0.f ? ev : NEG_SLOPE * ev;
    float m = funkey(mkey[dst * 4 + hd]);
    float alpha = __expf(ev - m) / (denom[dst * 4 + hd] + 1e-16f);
    float2 hv = *(const float2*)(hbuf + (size_t)src * 64 + 2 * p);
    atomicAdd(agg + (size_t)dst * 64 + 2 * p,     alpha * hv.x);
    atomicAdd(agg + (size_t)dst * 64 + 2 * p + 1, alpha * hv.y);
}

// --------------------------- K6: out GEMM + norm + LIF ----------------------
__global__ __launch_bounds__(256)
void k_out(const float* __restrict__ agg, const float* __restrict__ cnt,
           const float* __restrict__ P, const float* __restrict__ gamma,
           const float* __restrict__ beta, float* __restrict__ out) {
    int wave = (blockIdx.x * blockDim.x + threadIdx.x) >> 5;
    int lane = threadIdx.x & 31;
    int r0 = wave * 16;
    if (r0 >= NN) return;
    int half = lane >> 4;
    int l = lane & 15;

    float rc = 1.0f / fmaxf(cnt[r0 + l], 1.0f);       // mean aggregation
    const float* arow = agg + (size_t)(r0 + l) * 64 + 2 * half;
    v8f acc[4] = {v8f{}, v8f{}, v8f{}, v8f{}};

    for (int k = 0; k < 64; k += 4) {
        float2 av = *(const float2*)(arow + k);
        v2f a; a[0] = av.x * rc; a[1] = av.y * rc;
#pragma unroll
        for (int t = 0; t < 4; ++t) {
            float2 bv = *(const float2*)(P + (size_t)(t * 16 + l) * 64 + k + 2 * half);
            v2f b; b[0] = bv.x; b[1] = bv.y;
            acc[t] = __builtin_amdgcn_wmma_f32_16x16x4_f32(
                false, a, false, b, (short)0, acc[t], false, false);
        }
    }

    // per-row mean / variance over the 64 channels
    float s[8], q[8];
#pragma unroll
    for (int r = 0; r < 8; ++r) { s[r] = 0.f; q[r] = 0.f; }
#pragma unroll
    for (int t = 0; t < 4; ++t)
#pragma unroll
        for (int r = 0; r < 8; ++r) {
            float v = acc[t][r];
            s[r] += v;
            q[r] += v * v;
        }
#pragma unroll
    for (int r = 0; r < 8; ++r)
#pragma unroll
        for (int m = 1; m < 16; m <<= 1) {
            s[r] += __shfl_xor(s[r], m, 32);
            q[r] += __shfl_xor(q[r], m, 32);
        }

#pragma unroll
    for (int t = 0; t < 4; ++t) {
        float g  = gamma[t * 16 + l];
        float bt = beta[t * 16 + l];
#pragma unroll
        for (int r = 0; r < 8; ++r) {
            int row = r0 + half * 8 + r;
            float mean = s[r] * (1.0f / 64.0f);
            float var  = q[r] * (1.0f / 64.0f) - mean * mean;
            float zn = (acc[t][r] - mean) * __frsqrt_rn(var + EPS_N); // RHO*V_TH = 1
            zn = zn * g + bt;
            float v = zn * TAU_INV;                     // LIF membrane, v0 = 0
            out[(size_t)row * 64 + t * 16 + l] = (v >= V_TH) ? 1.0f : 0.0f;
        }
    }
}

// ---------------------------------------------------------------------------
extern "C" void kernel_launch(void* const* d_in, const int* in_sizes, int n_in,
                              void* d_out, int out_size, void* d_ws, size_t ws_size,
                              hipStream_t stream) {
    const float* x     = (const float*)d_in[0];
    const int*   ei    = (const int*)  d_in[1];   // edge_index, [2, NE]
    const float* W     = (const float*)d_in[2];   // proj_weight [64,64]
    const float* P     = (const float*)d_in[3];   // proj_out    [64,64]
    const float* att_s = (const float*)d_in[4];   // [1,4,16]
    const float* att_d = (const float*)d_in[5];
    const float* gamma = (const float*)d_in[6];
    const float* beta  = (const float*)d_in[7];
    float* out = (float*)d_out;
    float* ws  = (float*)d_ws;

    float*    hbuf  = ws + OFF_H;
    float*    asrc  = ws + OFF_ASRC;
    float*    adst  = ws + OFF_ADST;
    unsigned* mkey  = (unsigned*)(ws + OFF_MKEY);
    float*    denom = ws + OFF_DEN;
    float*    cnt   = ws + OFF_CNT;
    float*    agg   = ws + OFF_AGG;

    const int waves      = (NN + 15) / 16;                 // 6250
    const int gemm_grid  = (waves + 7) / 8;                // 8 waves / block
    const int init_grid  = ((OFF_END - OFF_MKEY) / 4 + 255) / 256;
    const int eh_grid    = (NE * 4 + 255) / 256;
    const int agg_grid   = (NE * 32 + 255) / 256;

    k_proj_att<<<gemm_grid, 256, 0, stream>>>(x, W, att_s, att_d, hbuf, asrc, adst);
    k_init    <<<init_grid, 256, 0, stream>>>(ws);
    k_edge_max<<<eh_grid,   256, 0, stream>>>(ei, asrc, adst, mkey, cnt);
    k_edge_den<<<eh_grid,   256, 0, stream>>>(ei, asrc, adst, mkey, denom);
    k_edge_agg<<<agg_grid,  256, 0, stream>>>(ei, asrc, adst, mkey, denom, hbuf, agg);
    k_out     <<<gemm_grid, 256, 0, stream>>>(agg, cnt, P, gamma, beta, out);
}